// LearnedRouter_2018634629284
// MI455X (gfx1250) — compile-verified
//
#include <hip/hip_runtime.h>
#include <math.h>

typedef __attribute__((ext_vector_type(2))) float v2f;
typedef __attribute__((ext_vector_type(4))) float v4f;
typedef __attribute__((ext_vector_type(8))) float v8f;

#define T_TOKENS 16384
#define H_DIM    4096
#define NE       8
#define BK       64      // K-chunk staged in LDS
#define XSTRIDE  68      // padded LDS row stride (floats): conflict-free A reads
#define SSTRIDE  20      // padded logit-park stride (floats)
#define NWAVES   8       // waves per block (256 threads, wave32)

__global__ __launch_bounds__(256, 4)
void router_wmma_kernel(const float* __restrict__ x,
                        const float* __restrict__ W,
                        float* __restrict__ out) {
  __shared__ float lds[NWAVES][16 * XSTRIDE];  // 34.8 KB total

  const int lane   = threadIdx.x & 31;
  const int wave   = threadIdx.x >> 5;
  const int lane16 = lane & 15;
  const int half   = lane >> 4;     // 0: K pair {0,1}, 1: K pair {2,3}
  const int erow   = lane & 7;      // expert row (N>=8 duplicates, unused)
  const int t0     = (blockIdx.x * NWAVES + wave) * 16;

  float* buf = lds[wave];
  const float* wrow = W + (size_t)erow * H_DIM + 2 * half;

  v8f acc = {0.f, 0.f, 0.f, 0.f, 0.f, 0.f, 0.f, 0.f};

  for (int kc = 0; kc < H_DIM; kc += BK) {
    // ---- stage 16x64 fp32 x-tile into LDS (coalesced b128 global loads) ----
#pragma unroll
    for (int it = 0; it < 8; ++it) {
      const int idx = it * 32 + lane;
      const int row = idx >> 4;          // 0..15
      const int col = (idx & 15) * 4;    // 0..60
      const v4f v = *reinterpret_cast<const v4f*>(
          x + (size_t)(t0 + row) * H_DIM + kc + col);
      *reinterpret_cast<v4f*>(&buf[row * XSTRIDE + col]) = v;
    }
    // speculative prefetch of next x-tile -> global_prefetch_b8
    if (kc + BK < H_DIM) {
      __builtin_prefetch(
          x + (size_t)(t0 + (lane >> 1)) * H_DIM + (kc + BK) + (lane & 1) * 32,
          0, 1);
    }
    __syncthreads();

    // ---- 16 WMMA k-steps over the staged tile ----
#pragma unroll
    for (int s = 0; s < 16; ++s) {
      // A fragment: x[t0 + L%16][kc + 4s + 2*half + {0,1}]  (ds_load_b64)
      const v2f a = *reinterpret_cast<const v2f*>(
          &buf[lane16 * XSTRIDE + 4 * s + 2 * half]);
      // B fragment: W[erow][kc + 4s + 2*half + {0,1}]       (cached b64)
      const v2f b = *reinterpret_cast<const v2f*>(wrow + kc + 4 * s);
      acc = __builtin_amdgcn_wmma_f32_16x16x4_f32(
          /*neg_a=*/false, a, /*neg_b=*/false, b,
          /*c_mod=*/(short)0, acc, /*reuse_a=*/false, /*reuse_b=*/false);
    }
    __syncthreads();
  }

  // ---- park 16x16 logits in LDS so each lane can gather one token's row ----
#pragma unroll
  for (int r = 0; r < 8; ++r)
    buf[(r + 8 * half) * SSTRIDE + lane16] = acc[r];
  __syncthreads();

  if (lane < 16) {
    const int t = t0 + lane16;
    float l[NE];
#pragma unroll
    for (int e = 0; e < NE; ++e) l[e] = buf[lane16 * SSTRIDE + e];

    // fp32 softmax over 8 experts
    float mx = l[0];
#pragma unroll
    for (int e = 1; e < NE; ++e) mx = fmaxf(mx, l[e]);
    float p[NE];
    float sum = 0.f;
#pragma unroll
    for (int e = 0; e < NE; ++e) { p[e] = expf(l[e] - mx); sum += p[e]; }
    const float inv = 1.0f / sum;
#pragma unroll
    for (int e = 0; e < NE; ++e) p[e] *= inv;

    // scores [T,8]
    float* srow = out + (size_t)t * NE;
    v4f s0 = {p[0], p[1], p[2], p[3]};
    v4f s1 = {p[4], p[5], p[6], p[7]};
    *reinterpret_cast<v4f*>(srow)     = s0;
    *reinterpret_cast<v4f*>(srow + 4) = s1;

    // top-2, first-occurrence tie-breaking (matches lax.top_k)
    int i0 = 0; float v0 = p[0];
#pragma unroll
    for (int e = 1; e < NE; ++e) if (p[e] > v0) { v0 = p[e]; i0 = e; }
    int i1 = (i0 == 0) ? 1 : 0; float v1 = p[i1];
#pragma unroll
    for (int e = 0; e < NE; ++e)
      if (e != i0 && p[e] > v1) { v1 = p[e]; i1 = e; }

    // expert_weights [T,2] then expert_indices [T,2] (int32 bits)
    float* wout = out + (size_t)T_TOKENS * NE + (size_t)t * 2;
    wout[0] = v0; wout[1] = v1;
    int* iout = reinterpret_cast<int*>(out) +
                (size_t)T_TOKENS * (NE + 2) + (size_t)t * 2;
    iout[0] = i0; iout[1] = i1;
  }
}

extern "C" void kernel_launch(void* const* d_in, const int* in_sizes, int n_in,
                              void* d_out, int out_size, void* d_ws, size_t ws_size,
                              hipStream_t stream) {
  (void)in_sizes; (void)n_in; (void)d_ws; (void)ws_size; (void)out_size;
  const float* x = (const float*)d_in[0];
  const float* W = (const float*)d_in[1];
  // d_in[2] is top_k == 2 (compile-time specialized)
  float* out = (float*)d_out;

  const int tiles_per_block = NWAVES;            // 16 tokens per wave
  const int grid = T_TOKENS / (16 * tiles_per_block);  // 128 blocks
  router_wmma_kernel<<<grid, 32 * NWAVES, 0, stream>>>(x, W, out);
}